// GATBotDetector_87651692577568
// MI455X (gfx1250) — compile-verified
//
#include <hip/hip_runtime.h>
#include <hip/hip_bf16.h>
#include <math.h>

typedef __attribute__((ext_vector_type(2))) float v2f;
typedef __attribute__((ext_vector_type(8))) float v8f;

// ---------------------------------------------------------------------------
// utility: grid-stride zero fill
// ---------------------------------------------------------------------------
__global__ void k_zero(float* __restrict__ p, long long n) {
  long long i = (long long)blockIdx.x * blockDim.x + threadIdx.x;
  long long stride = (long long)gridDim.x * blockDim.x;
  for (; i < n; i += stride) p[i] = 0.0f;
}

// ---------------------------------------------------------------------------
// WMMA fp32 GEMM:  out[m, j] = sum_k A[m,k] * W[j,k]   (A @ W^T)
//   A: [M,K] row-major, W: [Ncols,K] row-major.
//   grid = (ceil(M/(16*MT)), Ncols/128), block = 256 (8 waves, wave32).
//   Each block stages 16*MT rows x 128 K-chunk of A in padded LDS
//   (stride 132 -> A-frag rows land on distinct banks). Each wave owns a
//   16-column slab and keeps MT accumulators: one B float2 load feeds MT
//   v_wmma_f32_16x16x4_f32 ops targeting different accumulators, breaking
//   the WMMA RAW chain and cutting W traffic by MT.
// ---------------------------------------------------------------------------
template <int MT>
__global__ __launch_bounds__(256) void k_gemm_wmma(
    const float* __restrict__ A, const float* __restrict__ W,
    float* __restrict__ out, int M, int K, int Ncols)
{
  constexpr int KC = 128;               // K chunk staged in LDS
  constexpr int SP = KC + 4;            // padded row stride (floats)
  __shared__ float lds[16 * MT * SP];

  const int rowBase = blockIdx.x * (16 * MT);
  const int colBase = blockIdx.y * 128;
  const int tid = threadIdx.x;

  const int wave = tid >> 5;
  const int lane = tid & 31;
  const int m   = lane & 15;            // A-frag row within tile (ISA 7.12.2)
  const int kb  = (lane >> 4) << 1;     // K sub-offset: 0 (lanes 0-15) / 2 (16-31)
  const int col = colBase + (wave << 4) + (lane & 15);  // B-frag column
  const float* __restrict__ wrow = W + (size_t)col * K;

  v8f c[MT];
  #pragma unroll
  for (int t = 0; t < MT; ++t) c[t] = (v8f){0.f,0.f,0.f,0.f,0.f,0.f,0.f,0.f};

  for (int kc = 0; kc < K; kc += KC) {
    if (kc) __syncthreads();            // protect LDS reuse across chunks

    // cooperative float4 fill of the (16*MT) x KC A slab (zero-pad rows >= M)
    const int n4 = (16 * MT * KC) >> 2;
    for (int i4 = tid; i4 < n4; i4 += 256) {
      int idx = i4 << 2;
      int row = idx >> 7;               // idx / KC
      int cc  = idx & (KC - 1);
      float4 v = make_float4(0.f, 0.f, 0.f, 0.f);
      int grow = rowBase + row;
      if (grow < M) v = *(const float4*)(A + (size_t)grow * K + kc + cc);
      *(float4*)(&lds[row * SP + cc]) = v;
    }
    __syncthreads();

    #pragma unroll 2
    for (int k0 = 0; k0 < KC; k0 += 4) {
      v2f b;
      float2 bw = *(const float2*)(wrow + kc + k0 + kb);  // even -> 8B aligned
      b.x = bw.x; b.y = bw.y;
      #pragma unroll
      for (int t = 0; t < MT; ++t) {
        v2f a;
        const float* arow = &lds[(t * 16 + m) * SP + k0 + kb];
        a.x = arow[0];
        a.y = arow[1];
        c[t] = __builtin_amdgcn_wmma_f32_16x16x4_f32(
            /*neg_a=*/false, a, /*neg_b=*/false, b,
            /*c_mod=*/(short)0, c[t], /*reuse_a=*/false, /*reuse_b=*/false);
      }
    }
  }

  // C/D layout: VGPR i -> row i for lanes 0-15, row i+8 for lanes 16-31.
  const int rowOfs = (lane >> 4) << 3;
  #pragma unroll
  for (int t = 0; t < MT; ++t) {
    #pragma unroll
    for (int i = 0; i < 8; ++i) {
      int grow = rowBase + t * 16 + i + rowOfs;
      if (grow < M) out[(size_t)grow * Ncols + col] = c[t][i];
    }
  }
}

// ---------------------------------------------------------------------------
// attention coefficients: a_src[n,h] = dot(xw[n,h,:], att_src[h,:]) (C = 128)
// one thread per (n,h)
// ---------------------------------------------------------------------------
__global__ void k_attn_coef(const float* __restrict__ xw,
                            const float* __restrict__ atts,
                            const float* __restrict__ attd,
                            float* __restrict__ as_, float* __restrict__ ad_,
                            int NH, int H)
{
  int tid = blockIdx.x * blockDim.x + threadIdx.x;
  if (tid >= NH) return;
  int h = tid % H;
  const float4* row = (const float4*)(xw + (size_t)tid * 128);
  const float4* s = (const float4*)(atts + (size_t)h * 128);
  const float4* d = (const float4*)(attd + (size_t)h * 128);
  float accs = 0.f, accd = 0.f;
  #pragma unroll 8
  for (int i = 0; i < 32; ++i) {
    float4 v = row[i], sv = s[i], dv = d[i];
    accs += v.x * sv.x + v.y * sv.y + v.z * sv.z + v.w * sv.w;
    accd += v.x * dv.x + v.y * dv.y + v.z * dv.z + v.w * dv.w;
  }
  as_[tid] = accs;
  ad_[tid] = accd;
}

// monotone float<->uint mapping for atomicMax-based segment max
__device__ __forceinline__ unsigned enc_f(float f) {
  unsigned u = __float_as_uint(f);
  return (u & 0x80000000u) ? ~u : (u | 0x80000000u);
}
__device__ __forceinline__ float dec_f(unsigned k) {
  unsigned u = (k & 0x80000000u) ? (k ^ 0x80000000u) : ~k;
  return __uint_as_float(u);
}

// edge id -> (src,dst); ids >= E are self loops
__device__ __forceinline__ void edge_nodes(const int* __restrict__ ei, int eid,
                                           int E, int& src, int& dst) {
  if (eid < E) { src = ei[eid]; dst = ei[E + eid]; }
  else         { src = eid - E; dst = eid - E; }
}

// ---------------------------------------------------------------------------
// edge pass 1: e = leaky_relu(a_src[src]+a_dst[dst]); segment max via atomicMax
// ---------------------------------------------------------------------------
__global__ void k_edge_logits(const int* __restrict__ ei,
                              const float* __restrict__ as_,
                              const float* __restrict__ ad_,
                              float* __restrict__ e, unsigned* __restrict__ mkey,
                              int E, int Etot, int H)
{
  int tid = blockIdx.x * blockDim.x + threadIdx.x;
  if (tid >= Etot * H) return;
  int eid = tid / H;
  int h = tid - eid * H;
  int src, dst;
  edge_nodes(ei, eid, E, src, dst);
  float v = as_[src * H + h] + ad_[dst * H + h];
  v = v > 0.f ? v : 0.2f * v;
  e[tid] = v;
  atomicMax(&mkey[dst * H + h], enc_f(v));
}

// ---------------------------------------------------------------------------
// edge pass 2: ex = exp(e - max[dst]); segment sum via atomicAdd
// ---------------------------------------------------------------------------
__global__ void k_edge_expsum(const int* __restrict__ ei,
                              const unsigned* __restrict__ mkey,
                              float* __restrict__ e, float* __restrict__ denom,
                              int E, int Etot, int H)
{
  int tid = blockIdx.x * blockDim.x + threadIdx.x;
  if (tid >= Etot * H) return;
  int eid = tid / H;
  int h = tid - eid * H;
  int src, dst;
  edge_nodes(ei, eid, E, src, dst);
  float mx = dec_f(mkey[dst * H + h]);
  float x = __expf(e[tid] - mx);
  e[tid] = x;
  atomicAdd(&denom[dst * H + h], x);
}

// ---------------------------------------------------------------------------
// edge pass 3: agg[dst] += (ex/denom[dst]) * xw[src]    (C = 128 per head)
// one wave (32 lanes) per edge; one float4 per lane per head
// ---------------------------------------------------------------------------
__global__ void k_edge_scatter(const int* __restrict__ ei,
                               const float* __restrict__ xw,
                               const float* __restrict__ ex,
                               const float* __restrict__ denom,
                               float* __restrict__ agg,
                               int E, int Etot, int H)
{
  int gw = (blockIdx.x * blockDim.x + threadIdx.x) >> 5;
  int lane = threadIdx.x & 31;
  if (gw >= Etot) return;                 // uniform per wave
  int src, dst;
  edge_nodes(ei, gw, E, src, dst);
  const float4* xs = (const float4*)(xw + (size_t)src * H * 128);
  float* ao = agg + (size_t)dst * H * 128;
  for (int h = 0; h < H; ++h) {
    float alpha = ex[(size_t)gw * H + h] / denom[dst * H + h];
    float4 v = xs[h * 32 + lane];
    float* p = ao + h * 128 + (lane << 2);
    atomicAdd(p + 0, alpha * v.x);
    atomicAdd(p + 1, alpha * v.y);
    atomicAdd(p + 2, alpha * v.z);
    atomicAdd(p + 3, alpha * v.w);
  }
}

// ---------------------------------------------------------------------------
// in-place h = elu(h + bias[d])
// ---------------------------------------------------------------------------
__global__ void k_elu_bias(float* __restrict__ h, const float* __restrict__ bias,
                           long long n, int D)
{
  long long i = (long long)blockIdx.x * blockDim.x + threadIdx.x;
  long long stride = (long long)gridDim.x * blockDim.x;
  for (; i < n; i += stride) {
    float v = h[i] + bias[(int)(i % D)];
    h[i] = v > 0.f ? v : (__expf(v) - 1.f);
  }
}

// ---------------------------------------------------------------------------
// classifier: logits[n,:] = elu(agg2[n,:] + b2) @ Wc^T + bc   (HID=128, 2 cls)
// ---------------------------------------------------------------------------
__global__ void k_classify(const float* __restrict__ agg2,
                           const float* __restrict__ b2,
                           const float* __restrict__ Wc,
                           const float* __restrict__ bc,
                           float* __restrict__ out, int N)
{
  int n = blockIdx.x * blockDim.x + threadIdx.x;
  if (n >= N) return;
  const float4* row = (const float4*)(agg2 + (size_t)n * 128);
  const float4* bb = (const float4*)b2;
  const float4* w0 = (const float4*)Wc;
  const float4* w1 = (const float4*)(Wc + 128);
  float a0 = 0.f, a1 = 0.f;
  #pragma unroll 8
  for (int i = 0; i < 32; ++i) {
    float4 v = row[i], b = bb[i];
    float4 e;
    float t;
    t = v.x + b.x; e.x = t > 0.f ? t : (__expf(t) - 1.f);
    t = v.y + b.y; e.y = t > 0.f ? t : (__expf(t) - 1.f);
    t = v.z + b.z; e.z = t > 0.f ? t : (__expf(t) - 1.f);
    t = v.w + b.w; e.w = t > 0.f ? t : (__expf(t) - 1.f);
    float4 a = w0[i], c = w1[i];
    a0 += e.x * a.x + e.y * a.y + e.z * a.z + e.w * a.w;
    a1 += e.x * c.x + e.y * c.y + e.z * c.z + e.w * c.w;
  }
  out[(size_t)n * 2 + 0] = a0 + bc[0];
  out[(size_t)n * 2 + 1] = a1 + bc[1];
}

// ---------------------------------------------------------------------------
extern "C" void kernel_launch(void* const* d_in, const int* in_sizes, int n_in,
                              void* d_out, int out_size, void* d_ws, size_t ws_size,
                              hipStream_t stream) {
  const float* x      = (const float*)d_in[0];   // [N,256]
  const int*   ei     = (const int*)d_in[1];     // [2,E]
  const float* W1     = (const float*)d_in[2];   // [512,256]
  const float* att1s  = (const float*)d_in[3];   // [4,128]
  const float* att1d  = (const float*)d_in[4];   // [4,128]
  const float* b1     = (const float*)d_in[5];   // [512]
  const float* W2     = (const float*)d_in[6];   // [128,512]
  const float* att2s  = (const float*)d_in[7];   // [1,128]
  const float* att2d  = (const float*)d_in[8];   // [1,128]
  const float* b2     = (const float*)d_in[9];   // [128]
  const float* Wc     = (const float*)d_in[10];  // [2,128]
  const float* bc     = (const float*)d_in[11];  // [2]
  float* out = (float*)d_out;

  const int N = in_sizes[0] / 256;
  const int E = in_sizes[1] / 2;
  const int T = E + N;                  // edges incl. self loops

  // ---- workspace layout (floats). First region must be zeroed each call. ----
  float* ws = (float*)d_ws;
  size_t o = 0;
  float*    agg1   = ws + o; o += (size_t)N * 512;
  float*    agg2   = ws + o; o += (size_t)N * 128;
  float*    denom1 = ws + o; o += (size_t)N * 4;
  float*    denom2 = ws + o; o += (size_t)N;
  unsigned* mkey1  = (unsigned*)(ws + o); o += (size_t)N * 4;
  unsigned* mkey2  = (unsigned*)(ws + o); o += (size_t)N;
  const long long zeroCount = (long long)o;
  float* xw1 = ws + o; o += (size_t)N * 512;
  float* xw2 = ws + o; o += (size_t)N * 128;
  float* a1s = ws + o; o += (size_t)N * 4;
  float* a1d = ws + o; o += (size_t)N * 4;
  float* a2s = ws + o; o += (size_t)N;
  float* a2d = ws + o; o += (size_t)N;
  float* e1  = ws + o; o += (size_t)T * 4;
  float* e2  = ws + o; o += (size_t)T;
  (void)ws_size; (void)n_in; (void)out_size;

  // 0) zero aggregation buffers / denominators / max-keys
  k_zero<<<4096, 256, 0, stream>>>(ws, zeroCount);

  // 1) xw1 = x @ W1^T  -> [N, 512]
  {
    dim3 grid((N + 63) / 64, 512 / 128);
    k_gemm_wmma<4><<<grid, 256, 0, stream>>>(x, W1, xw1, N, 256, 512);
  }

  // 2) attention coefficients, layer 1 (H=4)
  k_attn_coef<<<(N * 4 + 255) / 256, 256, 0, stream>>>(xw1, att1s, att1d, a1s, a1d, N * 4, 4);

  // 3) edge softmax + scatter, layer 1
  k_edge_logits<<<(T * 4 + 255) / 256, 256, 0, stream>>>(ei, a1s, a1d, e1, mkey1, E, T, 4);
  k_edge_expsum<<<(T * 4 + 255) / 256, 256, 0, stream>>>(ei, mkey1, e1, denom1, E, T, 4);
  k_edge_scatter<<<(T + 7) / 8, 256, 0, stream>>>(ei, xw1, e1, denom1, agg1, E, T, 4);

  // 4) h1 = elu(agg1 + b1)  (in place)
  k_elu_bias<<<4096, 256, 0, stream>>>(agg1, b1, (long long)N * 512, 512);

  // 5) xw2 = h1 @ W2^T -> [N, 128]
  {
    dim3 grid((N + 63) / 64, 128 / 128);
    k_gemm_wmma<4><<<grid, 256, 0, stream>>>(agg1, W2, xw2, N, 512, 128);
  }

  // 6) attention coefficients, layer 2 (H=1)
  k_attn_coef<<<(N + 255) / 256, 256, 0, stream>>>(xw2, att2s, att2d, a2s, a2d, N, 1);

  // 7) edge softmax + scatter, layer 2
  k_edge_logits<<<(T + 255) / 256, 256, 0, stream>>>(ei, a2s, a2d, e2, mkey2, E, T, 1);
  k_edge_expsum<<<(T + 255) / 256, 256, 0, stream>>>(ei, mkey2, e2, denom2, E, T, 1);
  k_edge_scatter<<<(T + 7) / 8, 256, 0, stream>>>(ei, xw2, e2, denom2, agg2, E, T, 1);

  // 8) logits = elu(agg2 + b2) @ Wc^T + bc
  k_classify<<<(N + 255) / 256, 256, 0, stream>>>(agg2, b2, Wc, bc, out, N);
}